// ImprovedGATv2Regressor_1271310319972
// MI455X (gfx1250) — compile-verified
//
#include <hip/hip_runtime.h>

typedef __attribute__((ext_vector_type(2))) float v2f;
typedef __attribute__((ext_vector_type(8))) float v8f;

#define HID 128
#define NEG_SLOPE 0.2f
#define POOL_CHUNK 512

// ---------------- helpers: order-preserving float <-> int for atomicMax ----
__device__ __forceinline__ int encOrder(float f) {
    int i = __float_as_int(f);
    return i >= 0 ? i : (i ^ 0x7fffffff);
}
__device__ __forceinline__ float decOrder(int i) {
    return __int_as_float(i >= 0 ? i : (i ^ 0x7fffffff));
}

// ---------------- fill kernels ---------------------------------------------
__global__ void fill_f32(float* p, float v, int n) {
    int i = blockIdx.x * blockDim.x + threadIdx.x;
    if (i < n) p[i] = v;
}
__global__ void fill_i32(int* p, int v, int n) {
    int i = blockIdx.x * blockDim.x + threadIdx.x;
    if (i < n) p[i] = v;
}

// ---------------- WMMA f32 GEMM: Y[M,N] = X[M,K] @ W[N,K]^T + bias ---------
// One wave computes a 16x64 strip (4 accumulator tiles reuse one A fragment).
// A layout (16x4 f32, ISA 7.12.2): lanes 0-15 rows M, lanes 16-31 same rows;
// VGPR0 holds K = 0 (lo half) / 2 (hi half), VGPR1 holds K = 1 / 3.
// B (4x16) is symmetric: lane = column N, same K split.
// K-loop staged 4 WMMA-steps deep: all 20 b64 loads issued before consumption
// so the scheduler can overlap them instead of wait-0 per WMMA.
__global__ void gemm_bias_wmma(const float* __restrict__ X,
                               const float* __restrict__ W,
                               const float* __restrict__ bias,
                               float* __restrict__ Y,
                               int M, int N, int K, int act)
{
    const int lane = threadIdx.x & 31;
    const int wave = (blockIdx.x * blockDim.x + threadIdx.x) >> 5;
    const int tilesN = N >> 6;             // 64 columns per wave
    const int tilesM = M >> 4;
    if (wave >= tilesM * tilesN) return;
    const int tm = wave / tilesN;
    const int tn = wave % tilesN;
    const int half = lane >> 4;
    const int l = lane & 15;

    const float* xrow = X + (size_t)(tm * 16 + l) * K + 2 * half;
    const float* w0 = W + (size_t)(tn * 64 +  0 + l) * K + 2 * half;
    const float* w1 = W + (size_t)(tn * 64 + 16 + l) * K + 2 * half;
    const float* w2 = W + (size_t)(tn * 64 + 32 + l) * K + 2 * half;
    const float* w3 = W + (size_t)(tn * 64 + 48 + l) * K + 2 * half;

    v8f c0 = {}, c1 = {}, c2 = {}, c3 = {};
    for (int k = 0; k < K; k += 16) {      // 4 WMMA K-steps per iteration
        v2f a[4], b0[4], b1[4], b2[4], b3[4];
#pragma unroll
        for (int kk = 0; kk < 4; ++kk) a[kk]  = *(const v2f*)(xrow + k + 4 * kk);
#pragma unroll
        for (int kk = 0; kk < 4; ++kk) b0[kk] = *(const v2f*)(w0 + k + 4 * kk);
#pragma unroll
        for (int kk = 0; kk < 4; ++kk) b1[kk] = *(const v2f*)(w1 + k + 4 * kk);
#pragma unroll
        for (int kk = 0; kk < 4; ++kk) b2[kk] = *(const v2f*)(w2 + k + 4 * kk);
#pragma unroll
        for (int kk = 0; kk < 4; ++kk) b3[kk] = *(const v2f*)(w3 + k + 4 * kk);
#pragma unroll
        for (int kk = 0; kk < 4; ++kk) {
            c0 = __builtin_amdgcn_wmma_f32_16x16x4_f32(false, a[kk], false, b0[kk], (short)0, c0, false, false);
            c1 = __builtin_amdgcn_wmma_f32_16x16x4_f32(false, a[kk], false, b1[kk], (short)0, c1, false, false);
            c2 = __builtin_amdgcn_wmma_f32_16x16x4_f32(false, a[kk], false, b2[kk], (short)0, c2, false, false);
            c3 = __builtin_amdgcn_wmma_f32_16x16x4_f32(false, a[kk], false, b3[kk], (short)0, c3, false, false);
        }
    }

    const int colBase = tn * 64 + l;
    const float bs0 = bias[colBase +  0];
    const float bs1 = bias[colBase + 16];
    const float bs2 = bias[colBase + 32];
    const float bs3 = bias[colBase + 48];
    const int row0 = tm * 16 + half * 8;
#pragma unroll
    for (int r = 0; r < 8; ++r) {
        float* yp = Y + (size_t)(row0 + r) * N + colBase;
        float o0 = c0[r] + bs0;
        float o1 = c1[r] + bs1;
        float o2 = c2[r] + bs2;
        float o3 = c3[r] + bs3;
        if (act) { o0 = fmaxf(o0, 0.f); o1 = fmaxf(o1, 0.f);
                   o2 = fmaxf(o2, 0.f); o3 = fmaxf(o3, 0.f); }
        yp[0] = o0; yp[16] = o1; yp[32] = o2; yp[48] = o3;
    }
}

// ---------------- GAT edge kernels -----------------------------------------
// One wave per edge; lane owns channels {lane, lane+32, lane+64, lane+96}.
__global__ void edge_score(const float* __restrict__ xl, const float* __restrict__ xr,
                           const int* __restrict__ ei, const float* __restrict__ att,
                           float* __restrict__ p, int* __restrict__ smax,
                           int E, int Etot, int heads)
{
    const int lane = threadIdx.x & 31;
    const int e = (blockIdx.x * blockDim.x + threadIdx.x) >> 5;
    if (e >= Etot) return;
    int s, d;
    if (e < E) { s = ei[e]; d = ei[E + e]; } else { s = d = e - E; }
    const float* xs = xl + (size_t)s * HID;
    const float* xd = xr + (size_t)d * HID;
    float acc[4];
#pragma unroll
    for (int j = 0; j < 4; ++j) {
        int c = j * 32 + lane;
        float v = xs[c] + xd[c];
        v = v > 0.f ? v : v * NEG_SLOPE;
        acc[j] = v * att[c];   // att is (heads,outc) row-major; heads*outc==128
    }
#pragma unroll
    for (int j = 0; j < 4; ++j)
#pragma unroll
        for (int off = 16; off; off >>= 1)
            acc[j] += __shfl_xor(acc[j], off, 32);
    if (lane == 0) {
        if (heads == 4) {
#pragma unroll
            for (int h = 0; h < 4; ++h) {
                p[(size_t)e * 4 + h] = acc[h];
                atomicMax(&smax[d * 4 + h], encOrder(acc[h]));
            }
        } else {
            float sc = acc[0] + acc[1] + acc[2] + acc[3];
            p[e] = sc;
            atomicMax(&smax[d], encOrder(sc));
        }
    }
}

__global__ void edge_exp(const int* __restrict__ ei, float* __restrict__ p,
                         const int* __restrict__ smax, float* __restrict__ denom,
                         int E, int Etot, int heads)
{
    int idx = blockIdx.x * blockDim.x + threadIdx.x;
    if (idx >= Etot * heads) return;
    int e = idx / heads;
    int h = idx - e * heads;
    int d = (e < E) ? ei[E + e] : (e - E);
    float pe = __expf(p[idx] - decOrder(smax[d * heads + h]));
    p[idx] = pe;
    atomicAdd(&denom[d * heads + h], pe);
}

__global__ void edge_agg(const float* __restrict__ xl, const int* __restrict__ ei,
                         const float* __restrict__ p, const float* __restrict__ denom,
                         float* __restrict__ agg, int E, int Etot, int heads, int ocShift)
{
    const int lane = threadIdx.x & 31;
    const int e = (blockIdx.x * blockDim.x + threadIdx.x) >> 5;
    if (e >= Etot) return;
    int s, d;
    if (e < E) { s = ei[e]; d = ei[E + e]; } else { s = d = e - E; }
#pragma unroll
    for (int j = 0; j < 4; ++j) {
        int c = j * 32 + lane;
        int h = c >> ocShift;
        float alpha = p[(size_t)e * heads + h] / denom[d * heads + h];
        atomicAdd(&agg[(size_t)d * HID + c], xl[(size_t)s * HID + c] * alpha);
    }
}

// ---------------- LayerNorm + ReLU + residual (wave per node) --------------
__global__ void node_post(const float* __restrict__ agg, const float* __restrict__ gbias,
                          const float* __restrict__ g, const float* __restrict__ b,
                          float* __restrict__ h, int Nn, int residual)
{
    const int lane = threadIdx.x & 31;
    const int n = (blockIdx.x * blockDim.x + threadIdx.x) >> 5;
    if (n >= Nn) return;
    float v[4];
    float sum = 0.f;
#pragma unroll
    for (int j = 0; j < 4; ++j) {
        int c = j * 32 + lane;
        v[j] = agg[(size_t)n * HID + c] + gbias[c];
        sum += v[j];
    }
#pragma unroll
    for (int off = 16; off; off >>= 1) sum += __shfl_xor(sum, off, 32);
    float mean = sum * (1.0f / HID);
    float ssq = 0.f;
#pragma unroll
    for (int j = 0; j < 4; ++j) { float dv = v[j] - mean; ssq += dv * dv; }
#pragma unroll
    for (int off = 16; off; off >>= 1) ssq += __shfl_xor(ssq, off, 32);
    float rstd = rsqrtf(ssq * (1.0f / HID) + 1e-5f);
#pragma unroll
    for (int j = 0; j < 4; ++j) {
        int c = j * 32 + lane;
        float out = fmaxf((v[j] - mean) * rstd * g[c] + b[c], 0.f);
        size_t idx = (size_t)n * HID + c;
        h[idx] = residual ? (h[idx] + out) : out;
    }
}

// ---------------- gating + pooling -----------------------------------------
__global__ void gate_score(const float* __restrict__ ghid, const float* __restrict__ gW2,
                           const float* __restrict__ gb2, const int* __restrict__ batch,
                           float* __restrict__ gate, int* __restrict__ gmax, int Nn)
{
    const int lane = threadIdx.x & 31;
    const int n = (blockIdx.x * blockDim.x + threadIdx.x) >> 5;
    if (n >= Nn) return;
    float sum = 0.f;
#pragma unroll
    for (int j = 0; j < 4; ++j) {
        int c = j * 32 + lane;
        sum += ghid[(size_t)n * HID + c] * gW2[c];
    }
#pragma unroll
    for (int off = 16; off; off >>= 1) sum += __shfl_xor(sum, off, 32);
    if (lane == 0) {
        float sc = sum + gb2[0];
        gate[n] = sc;
        atomicMax(&gmax[batch[n]], encOrder(sc));
    }
}

__global__ void gate_w(const float* __restrict__ gate, const int* __restrict__ gmax,
                       const int* __restrict__ batch, float* __restrict__ w,
                       float* __restrict__ wsum, int Nn)
{
    int n = blockIdx.x * blockDim.x + threadIdx.x;
    if (n >= Nn) return;
    int g = batch[n];
    float we = __expf(gate[n] - decOrder(gmax[g]));
    w[n] = we;
    atomicAdd(&wsum[g], we);
}

// Each thread owns one channel column; per-block partials in LDS, then few atomics.
__global__ void pool_kernel(const float* __restrict__ h, const float* __restrict__ w,
                            const float* __restrict__ wsum, const int* __restrict__ batch,
                            float* __restrict__ pooled, int Nn)
{
    __shared__ float sdata[8 * HID];
    const int c = threadIdx.x;           // 0..127
#pragma unroll
    for (int g = 0; g < 8; ++g) sdata[g * HID + c] = 0.f;
    int n0 = blockIdx.x * POOL_CHUNK;
    int n1 = n0 + POOL_CHUNK; if (n1 > Nn) n1 = Nn;
    for (int n = n0; n < n1; ++n) {
        int g = batch[n];
        sdata[g * HID + c] += (w[n] / wsum[g]) * h[(size_t)n * HID + c];
    }
#pragma unroll
    for (int g = 0; g < 8; ++g)
        atomicAdd(&pooled[g * HID + c], sdata[g * HID + c]);
}

// ---------------- final tiny MLP (8x128 -> 8x64 -> 8) ----------------------
__global__ void mlp_kernel(const float* __restrict__ pooled,
                           const float* __restrict__ W1, const float* __restrict__ b1,
                           const float* __restrict__ W2, const float* __restrict__ b2,
                           const float* __restrict__ W3, const float* __restrict__ b3,
                           float* __restrict__ out)
{
    __shared__ float z1[8 * HID];
    __shared__ float z2[8 * 64];
    int t = threadIdx.x;                 // 128 threads
    for (int g = 0; g < 8; ++g) {
        float s = b1[t];
        for (int k = 0; k < HID; ++k) s += pooled[g * HID + k] * W1[t * HID + k];
        z1[g * HID + t] = fmaxf(s, 0.f);
    }
    __syncthreads();
    if (t < 64) {
        for (int g = 0; g < 8; ++g) {
            float s = b2[t];
            for (int k = 0; k < HID; ++k) s += z1[g * HID + k] * W2[t * HID + k];
            z2[g * 64 + t] = fmaxf(s, 0.f);
        }
    }
    __syncthreads();
    if (t < 8) {
        float s = b3[0];
        for (int k = 0; k < 64; ++k) s += z2[t * 64 + k] * W3[k];
        out[t] = s;
    }
}

// ---------------- host side -------------------------------------------------
extern "C" void kernel_launch(void* const* d_in, const int* in_sizes, int n_in,
                              void* d_out, int out_size, void* d_ws, size_t ws_size,
                              hipStream_t stream)
{
    (void)n_in; (void)out_size; (void)ws_size;
    const float* x     = (const float*)d_in[0];
    const int*   ei    = (const int*)d_in[1];
    const int*   batch = (const int*)d_in[2];
    const float* Wp    = (const float*)d_in[3];
    const float* bp    = (const float*)d_in[4];
    const float* Wl[3]    = {(const float*)d_in[5],  (const float*)d_in[6],  (const float*)d_in[7]};
    const float* bl[3]    = {(const float*)d_in[8],  (const float*)d_in[9],  (const float*)d_in[10]};
    const float* Wr[3]    = {(const float*)d_in[11], (const float*)d_in[12], (const float*)d_in[13]};
    const float* br[3]    = {(const float*)d_in[14], (const float*)d_in[15], (const float*)d_in[16]};
    const float* att[3]   = {(const float*)d_in[17], (const float*)d_in[18], (const float*)d_in[19]};
    const float* gbias[3] = {(const float*)d_in[20], (const float*)d_in[21], (const float*)d_in[22]};
    const float* lng[3]   = {(const float*)d_in[23], (const float*)d_in[24], (const float*)d_in[25]};
    const float* lnb[3]   = {(const float*)d_in[26], (const float*)d_in[27], (const float*)d_in[28]};
    const float* gW1 = (const float*)d_in[29];
    const float* gb1 = (const float*)d_in[30];
    const float* gW2 = (const float*)d_in[31];
    const float* gb2 = (const float*)d_in[32];
    const float* hW1 = (const float*)d_in[33];
    const float* hb1 = (const float*)d_in[34];
    const float* hW2 = (const float*)d_in[35];
    const float* hb2 = (const float*)d_in[36];
    const float* hW3 = (const float*)d_in[37];
    const float* hb3 = (const float*)d_in[38];

    const int Nn   = in_sizes[0] / 256;      // 50000
    const int E    = in_sizes[1] / 2;        // 800000
    const int K0   = in_sizes[3] / HID;      // 256
    const int Etot = E + Nn;                 // self-loops appended virtually
    const int headsArr[3]   = {4, 4, 1};
    const int ocShiftArr[3] = {5, 5, 7};     // log2(outc)

    // workspace layout
    char* wsb = (char*)d_ws;
    size_t off = 0;
    auto wsAlloc = [&](size_t bytes) -> void* {
        void* pp = wsb + off;
        off = (off + bytes + 255) & ~(size_t)255;
        return pp;
    };
    float* h      = (float*)wsAlloc((size_t)Nn * HID * 4);
    float* xl     = (float*)wsAlloc((size_t)Nn * HID * 4);
    float* xr     = (float*)wsAlloc((size_t)Nn * HID * 4);
    float* agg    = (float*)wsAlloc((size_t)Nn * HID * 4);
    float* p      = (float*)wsAlloc((size_t)Etot * 4 * 4);
    int*   smax   = (int*)  wsAlloc((size_t)Nn * 4 * 4);
    float* denom  = (float*)wsAlloc((size_t)Nn * 4 * 4);
    float* gate   = (float*)wsAlloc((size_t)Nn * 4);
    float* wbuf   = (float*)wsAlloc((size_t)Nn * 4);
    int*   gmax   = (int*)  wsAlloc(8 * 4);
    float* wsum   = (float*)wsAlloc(8 * 4);
    float* pooled = (float*)wsAlloc(8 * HID * 4);

    auto gemmLaunch = [&](const float* X, const float* W, const float* bias, float* Y,
                          int M, int N, int K, int act) {
        int waves = (M >> 4) * (N >> 6);
        int blocks = (waves * 32 + 255) / 256;
        gemm_bias_wmma<<<blocks, 256, 0, stream>>>(X, W, bias, Y, M, N, K, act);
    };

    // h = x @ Wp^T + bp
    gemmLaunch(x, Wp, bp, h, Nn, HID, K0, 0);

    const int waveBlocksE = (Etot * 32 + 255) / 256;
    const int waveBlocksN = (Nn * 32 + 255) / 256;

    for (int i = 0; i < 3; ++i) {
        const int hs = headsArr[i];
        gemmLaunch(h, Wl[i], bl[i], xl, Nn, HID, HID, 0);
        gemmLaunch(h, Wr[i], br[i], xr, Nn, HID, HID, 0);
        fill_f32<<<(Nn * HID + 255) / 256, 256, 0, stream>>>(agg, 0.f, Nn * HID);
        fill_f32<<<(Nn * hs + 255) / 256, 256, 0, stream>>>(denom, 0.f, Nn * hs);
        fill_i32<<<(Nn * hs + 255) / 256, 256, 0, stream>>>(smax, (int)0x80000000, Nn * hs);
        edge_score<<<waveBlocksE, 256, 0, stream>>>(xl, xr, ei, att[i], p, smax, E, Etot, hs);
        edge_exp<<<(Etot * hs + 255) / 256, 256, 0, stream>>>(ei, p, smax, denom, E, Etot, hs);
        edge_agg<<<waveBlocksE, 256, 0, stream>>>(xl, ei, p, denom, agg, E, Etot, hs, ocShiftArr[i]);
        node_post<<<waveBlocksN, 256, 0, stream>>>(agg, gbias[i], lng[i], lnb[i], h, Nn, i > 0);
    }

    // gating + pooling
    gemmLaunch(h, gW1, gb1, xl, Nn, HID, HID, 1);   // reuse xl as gate hidden
    fill_i32<<<1, 32, 0, stream>>>(gmax, (int)0x80000000, 8);
    fill_f32<<<1, 32, 0, stream>>>(wsum, 0.f, 8);
    fill_f32<<<(8 * HID + 255) / 256, 256, 0, stream>>>(pooled, 0.f, 8 * HID);
    gate_score<<<waveBlocksN, 256, 0, stream>>>(xl, gW2, gb2, batch, gate, gmax, Nn);
    gate_w<<<(Nn + 255) / 256, 256, 0, stream>>>(gate, gmax, batch, wbuf, wsum, Nn);
    pool_kernel<<<(Nn + POOL_CHUNK - 1) / POOL_CHUNK, HID, 0, stream>>>(h, wbuf, wsum, batch, pooled, Nn);
    mlp_kernel<<<1, HID, 0, stream>>>(pooled, hW1, hb1, hW2, hb2, hW3, hb3, (float*)d_out);
}